// SAGE_75617194213707
// MI455X (gfx1250) — compile-verified
//
#include <hip/hip_runtime.h>
#include <hip/hip_bf16.h>

typedef __attribute__((ext_vector_type(2))) float v2f;
typedef __attribute__((ext_vector_type(8))) float v8f;

#define NN   10000
#define EE   320000
#define FF   128
#define DD1  128
#define HH1  64

// ---------------------------------------------------------------------------
// fp32 WMMA 16x16 tile GEMM: C(16x16) = A(16xK) * B(Kx16), row-major A/B.
// Per ISA 7.12.2 (32-bit A 16x4): lanes 0-15 hold M=lane, VGPR0/1 = K{0,1};
// lanes 16-31 hold M=lane-16, VGPR0/1 = K{2,3}.  B mirrors (K rows across
// VGPRs/half-waves, N across lanes).  One wave (32 lanes), EXEC all ones.
// ---------------------------------------------------------------------------
__device__ __forceinline__ v8f wmma_f32_tile(const float* __restrict__ A, int lda,
                                             const float* __restrict__ B, int ldb,
                                             int K, int m_base, int n_base) {
    const int lane = threadIdx.x & 31;
    const int half = lane >> 4;   // 0: K pair {k,k+1}, 1: K pair {k+2,k+3}
    const int l15  = lane & 15;
    v8f acc = {0.f, 0.f, 0.f, 0.f, 0.f, 0.f, 0.f, 0.f};
    const float* arow = A + (size_t)(m_base + l15) * lda + 2 * half;
    const float* bcol = B + n_base + l15 + (size_t)(2 * half) * ldb;
    for (int k = 0; k < K; k += 4) {
        v2f a = *(const v2f*)(arow + k);                 // contiguous K pair
        v2f b;
        b.x = bcol[(size_t)k * ldb];                      // row k+2*half
        b.y = bcol[(size_t)k * ldb + ldb];                // row k+2*half+1
        acc = __builtin_amdgcn_wmma_f32_16x16x4_f32(
            /*neg_a=*/false, a, /*neg_b=*/false, b,
            /*c_mod=*/(short)0, acc, /*reuse_a=*/false, /*reuse_b=*/false);
    }
    return acc;
}

// xw = x @ W_gcn   [NN,FF] @ [FF,DD1]
__global__ void gemm_xw_kernel(const float* __restrict__ x,
                               const float* __restrict__ Wg,
                               float* __restrict__ xw) {
    const int m_base = blockIdx.y * 16;
    const int n_base = blockIdx.x * 16;
    v8f c = wmma_f32_tile(x, FF, Wg, DD1, FF, m_base, n_base);
    const int lane = threadIdx.x & 31;
    const int half = lane >> 4;
    const int l15  = lane & 15;
#pragma unroll
    for (int j = 0; j < 8; ++j)   // C layout: VGPR j -> M = j (+8 for hi half)
        xw[(size_t)(m_base + j + 8 * half) * DD1 + n_base + l15] = c[j];
}

// a1 = tanh(h @ W_fc1 + b_fc1)   [NN,DD1] @ [DD1,HH1]
__global__ void gemm_a1_kernel(const float* __restrict__ h,
                               const float* __restrict__ W1,
                               const float* __restrict__ b1,
                               float* __restrict__ a1) {
    const int m_base = blockIdx.y * 16;
    const int n_base = blockIdx.x * 16;
    v8f c = wmma_f32_tile(h, DD1, W1, HH1, DD1, m_base, n_base);
    const int lane = threadIdx.x & 31;
    const int half = lane >> 4;
    const int l15  = lane & 15;
    const float bias = b1[n_base + l15];
#pragma unroll
    for (int j = 0; j < 8; ++j)
        a1[(size_t)(m_base + j + 8 * half) * HH1 + n_base + l15] = tanhf(c[j] + bias);
}

// ---------------------------------------------------------------------------
// Graph preprocessing
// ---------------------------------------------------------------------------
__global__ void zero_init_kernel(float* deg, float* deg_l, int* has, float* acc) {
    int i = blockIdx.x * blockDim.x + threadIdx.x;
    if (i < NN) { deg[i] = 0.f; deg_l[i] = 0.f; has[i] = 0; }
    if (i < 136) acc[i] = 0.f;   // STS[4] + SAS[4] + gsum[128]
}

__global__ void edge_deg_kernel(const int* __restrict__ row, const int* __restrict__ col,
                                const float* __restrict__ w,
                                float* deg, float* deg_l, int* has) {
    int e = blockIdx.x * blockDim.x + threadIdx.x;
    if (e >= EE) return;
    int r = row[e], c = col[e];
    float we = w[e];
    atomicAdd(&deg[c], we);                 // GCN degree at col
    if (r == c) atomicOr(&has[r], 1);       // existing self-loop
    else        atomicAdd(&deg_l[r], we);   // Laplacian degree at row, loops removed
}

__global__ void node_dinv_kernel(const float* __restrict__ deg,
                                 const float* __restrict__ deg_l,
                                 const int* __restrict__ has,
                                 float* dinv, float* dinv_l) {
    int i = blockIdx.x * blockDim.x + threadIdx.x;
    if (i >= NN) return;
    float lw = has[i] ? 0.f : 1.f;          // add_remaining_self_loops fill=1
    float d  = deg[i] + lw;
    dinv[i]  = d  > 0.f ? rsqrtf(d)  : 0.f;
    float dl = deg_l[i];
    dinv_l[i] = dl > 0.f ? rsqrtf(dl) : 0.f;
}

// h[i][:] = b_gcn + (dinv[i]^2 * loop_w[i]) * xw[i][:]  (self-loop term fused)
__global__ void h_init_kernel(const float* __restrict__ xw,
                              const float* __restrict__ dinv,
                              const int* __restrict__ has,
                              const float* __restrict__ bg,
                              float* __restrict__ h) {
    int idx = blockIdx.x * blockDim.x + threadIdx.x;
    if (idx >= NN * 32) return;
    int i = idx >> 5;
    int d = (idx & 31) * 4;
    float lw = has[i] ? 0.f : 1.f;
    float di = dinv[i];
    float nl = lw * di * di;
    float4 xv = *(const float4*)(xw + (size_t)i * DD1 + d);
    float4 bv = *(const float4*)(bg + d);
    float4 hv = { bv.x + nl * xv.x, bv.y + nl * xv.y,
                  bv.z + nl * xv.z, bv.w + nl * xv.w };
    *(float4*)(h + (size_t)i * DD1 + d) = hv;
}

// One wave per edge: h[c][:] += dinv[r]*w*dinv[c] * xw[r][:], 4 floats/lane.
__global__ void scatter_kernel(const int* __restrict__ row, const int* __restrict__ col,
                               const float* __restrict__ w,
                               const float* __restrict__ dinv,
                               const float* __restrict__ xw,
                               float* __restrict__ h) {
    int gid    = blockIdx.x * blockDim.x + threadIdx.x;
    int wave   = gid >> 5;
    int lane   = gid & 31;
    int nwaves = (gridDim.x * blockDim.x) >> 5;
    int d = lane * 4;
    for (int e = wave; e < EE; e += nwaves) {
        int r = row[e], c = col[e];
        float nrm = dinv[r] * w[e] * dinv[c];
        float4 xv = *(const float4*)(xw + (size_t)r * DD1 + d);
        float* hp = h + (size_t)c * DD1 + d;
        atomicAdd(hp + 0, nrm * xv.x);
        atomicAdd(hp + 1, nrm * xv.y);
        atomicAdd(hp + 2, nrm * xv.z);
        atomicAdd(hp + 3, nrm * xv.w);
    }
}

// ---------------------------------------------------------------------------
// fc2 + softmax + reductions: S, STS = S^T S, gsum[d] = sum_i (s0+s1)*h[i][d]
// ---------------------------------------------------------------------------
__global__ void mlp2_kernel(const float* __restrict__ a1,
                            const float* __restrict__ h,
                            const float* __restrict__ W2,
                            const float* __restrict__ b2,
                            float* __restrict__ S,
                            float* __restrict__ STS,    // acc[0..3]
                            float* __restrict__ gsum) { // acc[8..135]
    __shared__ float lg[DD1];
    __shared__ float lsts[4];
    int t = threadIdx.x;
    if (t < DD1) lg[t] = 0.f;
    if (t < 4)   lsts[t] = 0.f;
    __syncthreads();

    int i = blockIdx.x * blockDim.x + t;
    if (i < NN) {
        float z0 = b2[0], z1 = b2[1];
        const float* ar = a1 + (size_t)i * HH1;
#pragma unroll 8
        for (int j = 0; j < HH1; ++j) {
            float a = ar[j];
            z0 = fmaf(a, W2[j * 2 + 0], z0);
            z1 = fmaf(a, W2[j * 2 + 1], z1);
        }
        float m  = fmaxf(z0, z1);
        float e0 = expf(z0 - m), e1 = expf(z1 - m);
        float inv = 1.f / (e0 + e1);
        float s0 = e0 * inv, s1 = e1 * inv;
        S[i * 2 + 0] = s0;
        S[i * 2 + 1] = s1;
        atomicAdd(&lsts[0], s0 * s0);
        atomicAdd(&lsts[1], s0 * s1);
        atomicAdd(&lsts[2], s1 * s0);
        atomicAdd(&lsts[3], s1 * s1);
        float st = s0 + s1;
        const float* hr = h + (size_t)i * DD1;
        for (int d = 0; d < DD1; ++d)
            atomicAdd(&lg[d], st * hr[d]);
    }
    __syncthreads();
    if (t < DD1) atomicAdd(&gsum[t], lg[t]);
    if (t < 4)   atomicAdd(&STS[t], lsts[t]);
}

// SAS = S^T * Adj_norm * S, Adj_norm[r,c] = dinv_l[r]*w*dinv_l[c] (loops removed)
__global__ void sas_kernel(const int* __restrict__ row, const int* __restrict__ col,
                           const float* __restrict__ w,
                           const float* __restrict__ dinv_l,
                           const float* __restrict__ S,
                           float* __restrict__ SAS) {  // acc[4..7]
    __shared__ float ls[4];
    int t = threadIdx.x;
    if (t < 4) ls[t] = 0.f;
    __syncthreads();
    int e = blockIdx.x * blockDim.x + t;
    if (e < EE) {
        int r = row[e], c = col[e];
        if (r != c) {
            float wn = dinv_l[r] * w[e] * dinv_l[c];
            float a0 = S[r * 2], a1v = S[r * 2 + 1];
            float b0 = S[c * 2], b1v = S[c * 2 + 1];
            atomicAdd(&ls[0], wn * a0 * b0);
            atomicAdd(&ls[1], wn * a0 * b1v);
            atomicAdd(&ls[2], wn * a1v * b0);
            atomicAdd(&ls[3], wn * a1v * b1v);
        }
    }
    __syncthreads();
    if (t < 4) atomicAdd(&SAS[t], ls[t]);
}

__global__ void finalize_kernel(const float* __restrict__ acc,  // STS[4] SAS[4] gsum[128]
                                float* __restrict__ out) {
    int t = threadIdx.x;
    if (t < DD1) out[t] = 0.5f * acc[8 + t];   // mean(S^T h, axis=0)
    if (t == 0) {
        float A00 = acc[0] - acc[4], A01 = acc[1] - acc[5];
        float A10 = acc[2] - acc[6], A11 = acc[3] - acc[7];
        float r0 = fmaxf(fabsf(A00) + fabsf(A01), 1e-12f);
        float r1 = fmaxf(fabsf(A10) + fabsf(A11), 1e-12f);
        float d0 = A00 / r0 - 1.f;
        float d1 = A11 / r1 - 1.f;
        out[DD1] = 0.5f * (d0 * d0 + d1 * d1);
    }
}

// ---------------------------------------------------------------------------
extern "C" void kernel_launch(void* const* d_in, const int* in_sizes, int n_in,
                              void* d_out, int out_size, void* d_ws, size_t ws_size,
                              hipStream_t stream) {
    const float* x   = (const float*)d_in[0];
    const int*   ei  = (const int*)  d_in[1];
    const float* w   = (const float*)d_in[2];
    const float* Wg  = (const float*)d_in[3];
    const float* bg  = (const float*)d_in[4];
    const float* W1  = (const float*)d_in[5];
    const float* b1  = (const float*)d_in[6];
    const float* W2  = (const float*)d_in[7];
    const float* b2  = (const float*)d_in[8];
    const int* row = ei;
    const int* col = ei + EE;
    float* out = (float*)d_out;

    char* ws = (char*)d_ws;
    size_t off = 0;
    auto carve = [&](size_t bytes) -> void* {
        void* p = ws + off;
        off = (off + bytes + 255) & ~(size_t)255;
        return p;
    };
    float* deg    = (float*)carve(NN * sizeof(float));
    float* dinv   = (float*)carve(NN * sizeof(float));
    float* deg_l  = (float*)carve(NN * sizeof(float));
    float* dinv_l = (float*)carve(NN * sizeof(float));
    int*   has    = (int*)  carve(NN * sizeof(int));
    float* xw     = (float*)carve((size_t)NN * DD1 * sizeof(float));
    float* h      = (float*)carve((size_t)NN * DD1 * sizeof(float));
    float* a1     = (float*)carve((size_t)NN * HH1 * sizeof(float));
    float* S      = (float*)carve((size_t)NN * 2 * sizeof(float));
    float* acc    = (float*)carve(136 * sizeof(float));  // STS[4] SAS[4] gsum[128]

    // 1. xw = x @ W_gcn (WMMA f32)
    gemm_xw_kernel<<<dim3(DD1 / 16, NN / 16), 32, 0, stream>>>(x, Wg, xw);
    // 2. zero accumulators
    zero_init_kernel<<<(NN + 255) / 256, 256, 0, stream>>>(deg, deg_l, has, acc);
    // 3. degrees + self-loop flags
    edge_deg_kernel<<<(EE + 255) / 256, 256, 0, stream>>>(row, col, w, deg, deg_l, has);
    // 4. inverse sqrt degrees
    node_dinv_kernel<<<(NN + 255) / 256, 256, 0, stream>>>(deg, deg_l, has, dinv, dinv_l);
    // 5. h = b_gcn + self-loop term
    h_init_kernel<<<(NN * 32 + 255) / 256, 256, 0, stream>>>(xw, dinv, has, bg, h);
    // 6. edge scatter (dominant; ~330 MB of HBM traffic + f32 atomics)
    scatter_kernel<<<1024, 256, 0, stream>>>(row, col, w, dinv, xw, h);
    // 7. a1 = tanh(h @ W_fc1 + b1) (WMMA f32)
    gemm_a1_kernel<<<dim3(HH1 / 16, NN / 16), 32, 0, stream>>>(h, W1, b1, a1);
    // 8. fc2 + softmax + S^T S + gsum
    mlp2_kernel<<<(NN + 255) / 256, 256, 0, stream>>>(a1, h, W2, b2, S, acc, acc + 8);
    // 9. S^T Adj S
    sas_kernel<<<(EE + 255) / 256, 256, 0, stream>>>(row, col, w, dinv_l, S, acc + 4);
    // 10. pooled outputs
    finalize_kernel<<<1, 256, 0, stream>>>(acc, out);
}